// MessagePassing_2826088481288
// MI455X (gfx1250) — compile-verified
//
#include <hip/hip_runtime.h>

// CDNA5 (gfx1250) message-passing: gather + edge-scale + segmented scatter-sum.
// Segment-sum of each 16-edge tile is computed as D = A(16x16 one-hot*ev) x M(16x64)
// using V_WMMA_F32_16X16X4_F32 (4 K-chunks x 4 feature blocks = 16 wmma / tile),
// then only the distinct-segment rows are atomically added to out[].

typedef float v2f __attribute__((ext_vector_type(2)));
typedef float v8f __attribute__((ext_vector_type(8)));

#define D_FEAT 64
#define TILE_E 16

__global__ void mp_zero_kernel(float* __restrict__ out, int n) {
    int i = blockIdx.x * blockDim.x + threadIdx.x;
    if (i < n) out[i] = 0.0f;
}

__launch_bounds__(256)
__global__ void mp_wmma_kernel(const float* __restrict__ x,
                               const float* __restrict__ evals,
                               const int*   __restrict__ tgt,
                               const int*   __restrict__ src,
                               float*       __restrict__ out,
                               int n_edges, int n_tiles) {
    __shared__ int slot_tgt[8][16];          // per-wave slot -> global target row

    const int lane = threadIdx.x & 31;
    const int wid  = threadIdx.x >> 5;
    const int tile = blockIdx.x * 8 + wid;   // one 16-edge tile per wave32
    if (tile >= n_tiles) return;             // uniform per wave: EXEC stays full

    const int base = tile * TILE_E;

    // Prefetch next tile-group's index data into cache (global_prefetch_b8).
    __builtin_prefetch(&tgt[base + 8 * TILE_E], 0, 1);
    __builtin_prefetch(&src[base + 8 * TILE_E], 0, 1);

    // ---- per-lane edge data: lanes 0..15 own edges base+lane (clamped tail) ----
    int e = base + (lane & 15);
    if (e >= n_edges) e = n_edges - 1;       // clamp: tail lanes get ev=0 below
    const bool real  = (base + (lane & 15)) < n_edges;
    int   t_  = tgt[e];
    int   s_  = src[e];
    float ev_ = real ? evals[e] : 0.0f;

    // ---- segment heads among the 16 sorted targets of this tile ----
    int  t_prev = __shfl_up(t_, 1, 32);
    bool head   = (lane < 16) && ((lane == 0) || (t_ != t_prev));
    unsigned hb = __builtin_amdgcn_ballot_w32(head) & 0xFFFFu;
    // rank of this edge's segment within the tile
    int slotv  = __popc(hb & ((2u << lane) - 1u)) - 1;
    int nslots = __popc(hb);                 // uniform across the wave

    if (head) slot_tgt[wid][slotv] = t_;     // per-wave LDS; in-order per wave

    // ---- P = A x M via v_wmma_f32_16x16x4_f32 ----
    // A layout: lane = m (0..15) x2 K-halves; K = 2*(lane>>4) + vgpr
    // B layout: lane = n (0..15) x2 K-halves; same K mapping
    v8f acc0 = {0.f,0.f,0.f,0.f,0.f,0.f,0.f,0.f};
    v8f acc1 = acc0, acc2 = acc0, acc3 = acc0;
    const int m  = lane & 15;                // A row / B feature column
    const int kh = lane >> 4;                // K-half selector

#pragma unroll
    for (int c = 0; c < 4; ++c) {            // K-chunks of 4 edges
        const int l0 = c * 4 + kh * 2;       // local edge for K = kh*2 + 0
        int   sA0 = __shfl(slotv, l0,     32);
        int   sA1 = __shfl(slotv, l0 + 1, 32);
        float eA0 = __shfl(ev_,   l0,     32);
        float eA1 = __shfl(ev_,   l0 + 1, 32);
        v2f a;
        a.x = (sA0 == m) ? eA0 : 0.0f;       // scaled one-hot selection matrix
        a.y = (sA1 == m) ? eA1 : 0.0f;

        long r0 = (long)__shfl(s_, l0,     32) * D_FEAT;
        long r1 = (long)__shfl(s_, l0 + 1, 32) * D_FEAT;

        // M rows gathered directly from global: lanes 0..15 read 64B contiguous
        v2f b0, b1, b2, b3;
        b0.x = x[r0 +  0 + m];  b0.y = x[r1 +  0 + m];
        b1.x = x[r0 + 16 + m];  b1.y = x[r1 + 16 + m];
        b2.x = x[r0 + 32 + m];  b2.y = x[r1 + 32 + m];
        b3.x = x[r0 + 48 + m];  b3.y = x[r1 + 48 + m];

        acc0 = __builtin_amdgcn_wmma_f32_16x16x4_f32(false, a, false, b0, (short)0, acc0, false, false);
        acc1 = __builtin_amdgcn_wmma_f32_16x16x4_f32(false, a, false, b1, (short)0, acc1, false, false);
        acc2 = __builtin_amdgcn_wmma_f32_16x16x4_f32(false, a, false, b2, (short)0, acc2, false, false);
        acc3 = __builtin_amdgcn_wmma_f32_16x16x4_f32(false, a, false, b3, (short)0, acc3, false, false);
    }

    // ---- scatter distinct-segment rows only (sorted targets => few atomics) ----
    // D layout: VGPR v holds row = v + 8*(lane>=16), col = lane%16
    const int rbase = kh * 8;
#pragma unroll
    for (int v = 0; v < 8; ++v) {
        int row = rbase + v;
        if (row < nslots) {                  // rows >= nslots are all-zero
            long o = (long)slot_tgt[wid][row] * D_FEAT + m;
            atomicAdd(&out[o +  0], acc0[v]);
            atomicAdd(&out[o + 16], acc1[v]);
            atomicAdd(&out[o + 32], acc2[v]);
            atomicAdd(&out[o + 48], acc3[v]);
        }
    }
}

extern "C" void kernel_launch(void* const* d_in, const int* in_sizes, int n_in,
                              void* d_out, int out_size, void* d_ws, size_t ws_size,
                              hipStream_t stream) {
    const float* x   = (const float*)d_in[0];
    const float* ev  = (const float*)d_in[1];
    const int*   tgt = (const int*)d_in[2];
    const int*   src = (const int*)d_in[3];
    float*       out = (float*)d_out;

    const int n_edges = in_sizes[1];

    // d_out is poisoned; we accumulate with atomics, so zero it first.
    mp_zero_kernel<<<(out_size + 255) / 256, 256, 0, stream>>>(out, out_size);

    const int n_tiles  = (n_edges + TILE_E - 1) / TILE_E;   // 100000 for reference
    const int n_blocks = (n_tiles + 7) / 8;                 // 8 waves per block
    mp_wmma_kernel<<<n_blocks, 256, 0, stream>>>(x, ev, tgt, src, out, n_edges, n_tiles);
}